// ExtendedContrastiveLoss_41154376630491
// MI455X (gfx1250) — compile-verified
//
#include <hip/hip_runtime.h>
#include <math.h>

typedef float v2f __attribute__((ext_vector_type(2)));
typedef float v8f __attribute__((ext_vector_type(8)));

#define C_CLS   40
#define CPAD    48
#define EMB     32
#define PIX     (512*512)
#define BATCH   4

// workspace layout per sample (float offsets)
#define WS_SUMS   0        // 40*32
#define WS_CNT    1280     // 40
#define WS_HV     1320     // 40
#define WS_INTER  1360     // 40
#define WS_DENP   1400     // 40
#define WS_PUSH   1440     // 1
#define WS_MU     1444     // 48*32
#define WS_MSUM   2980     // 48
#define WS_STRIDE 3072

// 1/TWO_SIGMA = (-ln 0.9) / DELTA_VAR^2
#define INV_TS (0.10536051565782630f / 0.25f)

__device__ __forceinline__ float fast_sqrtf(float x) {
#if defined(__has_builtin)
#if __has_builtin(__builtin_amdgcn_sqrtf)
    return __builtin_amdgcn_sqrtf(x);   // raw v_sqrt_f32, ~1 ulp
#else
    return sqrtf(x);
#endif
#else
    return sqrtf(x);
#endif
}

__global__ void k_zero(float* ws, int n) {
    int i = blockIdx.x * 256 + threadIdx.x;
    if (i < n) ws[i] = 0.0f;
}

// ---------------- pass A: per-class sums & counts ----------------
#define PIXB 1024
__global__ void k_sums(const float* __restrict__ in, const int* __restrict__ tgt,
                       float* __restrict__ ws) {
    __shared__ float lsum[C_CLS * 33];   // stride 33: spread banks
    __shared__ float lcnt[C_CLS];
    __shared__ int   lt[PIXB];
    const int tid = threadIdx.x;
    const int bps = PIX / PIXB;                 // 256 blocks per sample
    const int b   = blockIdx.x / bps;
    const int pb  = (blockIdx.x % bps) * PIXB;

    for (int i = tid; i < C_CLS * 33; i += 256) lsum[i] = 0.0f;
    for (int i = tid; i < C_CLS;      i += 256) lcnt[i] = 0.0f;
    for (int i = tid; i < PIXB;       i += 256) lt[i] = tgt[(size_t)b * PIX + pb + i];
    __syncthreads();

    const float* inb = in + (size_t)b * EMB * PIX + pb;
    for (int c = 0; c < EMB; ++c) {
        const float* row = inb + (size_t)c * PIX;
        for (int j = tid; j < PIXB; j += 256) {
            float v  = row[j];                  // coalesced
            int  cls = lt[j];
            atomicAdd(&lsum[cls * 33 + c], v);
            if (c == 0) atomicAdd(&lcnt[cls], 1.0f);
        }
    }
    __syncthreads();

    float* wss = ws + (size_t)b * WS_STRIDE;
    for (int i = tid; i < C_CLS * EMB; i += 256) {
        int cls = i >> 5, c = i & 31;
        atomicAdd(&wss[WS_SUMS + i], lsum[cls * 33 + c]);
    }
    for (int i = tid; i < C_CLS; i += 256) atomicAdd(&wss[WS_CNT + i], lcnt[i]);
}

// ---------------- means + |mu|^2 (padded to 48 clusters) ----------------
__global__ void k_means(float* __restrict__ ws) {
    const int b = blockIdx.x, tid = threadIdx.x;
    float* wss = ws + (size_t)b * WS_STRIDE;
    for (int i = tid; i < CPAD * EMB; i += 256) {
        int cls = i >> 5, e = i & 31;
        float m = 0.0f;
        if (cls < C_CLS) {
            float cnt = fmaxf(wss[WS_CNT + cls], 1.0f);
            m = wss[WS_SUMS + cls * EMB + e] / cnt;
        }
        wss[WS_MU + i] = m;
    }
    __syncthreads();
    for (int i = tid; i < CPAD; i += 256) {
        float s = 0.0f;
        for (int e = 0; e < EMB; ++e) { float m = wss[WS_MU + i * EMB + e]; s += m * m; }
        wss[WS_MSUM + i] = s;
    }
}

// ---------------- fused WMMA distance + reductions ----------------
// Augmented GEMM: K = 36 (9 steps of 4).
//   A[p][k<32]  = e_p[k]        B[k<32][cl] = -2*mu_cl[k]
//   A[p][32]    = |e_p|^2       B[32][cl]   = 1
//   A[p][33]    = 1             B[33][cl]   = |mu_cl|^2
//   A[p][34,35] = 0             B[34,35]    = 0
// => accumulator = d2(p, cl) directly.
#define WAVES    8
#define TILE_PIX 64
#define BLK_PIX  (WAVES * TILE_PIX)   // 512 pixels / block

__global__ void k_main(const float* __restrict__ in, const int* __restrict__ tgt,
                       float* __restrict__ ws) {
    __shared__ float lmu[CPAD * 33];     // padded means, stride 33
    __shared__ float lmsum[CPAD];
    __shared__ float lhv[C_CLS], linter[C_CLS], ldenp[C_CLS];
    __shared__ float lpush;

    const int tid = threadIdx.x;
    const int bps = PIX / BLK_PIX;               // 512 blocks per sample
    const int b    = blockIdx.x / bps;
    const int pblk = (blockIdx.x % bps) * BLK_PIX;
    float* wss = ws + (size_t)b * WS_STRIDE;

    for (int i = tid; i < CPAD * EMB; i += 256) {
        int cls = i >> 5, e = i & 31;
        lmu[cls * 33 + e] = wss[WS_MU + i];
    }
    for (int i = tid; i < CPAD;  i += 256) lmsum[i] = wss[WS_MSUM + i];
    for (int i = tid; i < C_CLS; i += 256) { lhv[i] = 0.f; linter[i] = 0.f; ldenp[i] = 0.f; }
    if (tid == 0) lpush = 0.0f;
    __syncthreads();

    const int wave = tid >> 5;
    const int lane = tid & 31;
    const int lm   = lane & 15;
    const int hi   = lane >> 4;
    const int pbase = pblk + wave * TILE_PIX;

    const float* inb = in  + (size_t)b * EMB * PIX;
    const int*   tb  = tgt + (size_t)b * PIX;

    // Hoisted B fragments (loop-invariant over pixels), scaled by -2, plus the
    // augmentation step k=8: lanes 0-15 -> {1, msum}, lanes 16-31 -> {0, 0}.
    v2f bfrag[3][9];
    #pragma unroll
    for (int n = 0; n < 3; ++n) {
        const int cl = lm + n * 16;
        #pragma unroll
        for (int k = 0; k < 8; ++k) {
            int K0 = 4 * k + 2 * hi;
            bfrag[n][k].x = -2.0f * lmu[cl * 33 + K0];
            bfrag[n][k].y = -2.0f * lmu[cl * 33 + K0 + 1];
        }
        bfrag[n][8].x = hi ? 0.0f : 1.0f;
        bfrag[n][8].y = hi ? 0.0f : lmsum[cl];
    }

    float hvA[3]   = {0.f, 0.f, 0.f};
    float intA[3]  = {0.f, 0.f, 0.f};
    float denA[3]  = {0.f, 0.f, 0.f};
    float pushA[3] = {0.f, 0.f, 0.f};

    for (int m = 0; m < 4; ++m) {
        const int ppix = pbase + m * 16 + lm;
        // A fragments: 16x4 f32 layout — lanes 0-15: K=4k,4k+1 ; lanes 16-31: K=4k+2,4k+3
        v2f a[9];
        float eep = 0.0f;
        #pragma unroll
        for (int k = 0; k < 8; ++k) {
            int K0 = 4 * k + 2 * hi;
            float x = inb[(size_t)K0 * PIX + ppix];
            float y = inb[(size_t)(K0 + 1) * PIX + ppix];
            a[k].x = x; a[k].y = y;
            eep += x * x + y * y;
        }
        float ee = eep + __shfl_xor(eep, 16, 32);   // full |e|^2 for pixel lm (lands in-place)
        a[8].x = hi ? 0.0f : ee;
        a[8].y = hi ? 0.0f : 1.0f;

        int tm = tb[ppix];                          // class of pixel lm
        int tq[8];
        #pragma unroll
        for (int i = 0; i < 8; ++i)
            tq[i] = __shfl(tm, i + 8 * hi, 32);     // class of the row this C vgpr holds

        #pragma unroll
        for (int n = 0; n < 3; ++n) {
            const int cl = lm + n * 16;
            v8f acc = {0.f, 0.f, 0.f, 0.f, 0.f, 0.f, 0.f, 0.f};
            #pragma unroll
            for (int k = 0; k < 9; ++k) {
                acc = __builtin_amdgcn_wmma_f32_16x16x4_f32(
                        false, a[k], false, bfrag[n][k], (short)0, acc, false, false);
            }
            // branchless elementwise epilogue: acc[i] == d2(p_row_i, cl)
            #pragma unroll
            for (int i = 0; i < 8; ++i) {
                float d2 = fmaxf(acc[i], 0.0f);
                float D  = fast_sqrtf(d2 + 1e-12f);
                float pm = __expf(-d2 * INV_TS);
                float dh = fmaxf(D - 0.5f, 0.0f);
                float ph = fmaxf(2.0f - D, 0.0f);
                bool own = (tq[i] == cl);
                bool bg  = (tq[i] == 0);
                hvA[n]   += own ? dh * dh : 0.0f;   // only true when cl<40 (tq<40)
                intA[n]  += own ? pm : 0.0f;
                denA[n]  += pm * pm;                // gated at reduction (cl in [1,40))
                pushA[n] += bg ? ph * ph : 0.0f;    // gated at reduction (cl in [1,40))
            }
        }
    }

    // lane->wave reduction: cluster fixed per (lane, n); halves share a cluster
    float pushAcc = 0.0f;
    #pragma unroll
    for (int n = 0; n < 3; ++n) {
        int cl = lm + n * 16;
        float lanemask = (cl >= 1 && cl < C_CLS) ? 1.0f : 0.0f;
        pushAcc += pushA[n] * lanemask;
        float h  = hvA[n]  + __shfl_xor(hvA[n],  16, 32);
        float it = intA[n] + __shfl_xor(intA[n], 16, 32);
        float dp = denA[n] + __shfl_xor(denA[n], 16, 32);
        if (hi == 0 && cl < C_CLS) {
            atomicAdd(&lhv[cl], h);
            if (cl >= 1) { atomicAdd(&linter[cl], it); atomicAdd(&ldenp[cl], dp); }
        }
    }
    for (int off = 16; off >= 1; off >>= 1) pushAcc += __shfl_xor(pushAcc, off, 32);
    if (lane == 0) atomicAdd(&lpush, pushAcc);
    __syncthreads();

    for (int i = tid; i < C_CLS; i += 256) {
        atomicAdd(&wss[WS_HV + i],    lhv[i]);
        atomicAdd(&wss[WS_INTER + i], linter[i]);
        atomicAdd(&wss[WS_DENP + i],  ldenp[i]);
    }
    if (tid == 0) atomicAdd(&wss[WS_PUSH], lpush);
}

// ---------------- finalize scalar loss ----------------
__global__ void k_final(const float* __restrict__ ws, float* __restrict__ out) {
    __shared__ float red[5];
    __shared__ float total;
    const int tid = threadIdx.x;
    if (tid == 0) total = 0.0f;
    for (int b = 0; b < BATCH; ++b) {
        if (tid < 5) red[tid] = 0.0f;
        __syncthreads();
        const float* wss = ws + (size_t)b * WS_STRIDE;

        float vpart = 0.f, dicep = 0.f, regp = 0.f;
        for (int c = tid; c < C_CLS; c += 256) {
            float cnt = fmaxf(wss[WS_CNT + c], 1.0f);
            if (c >= 1) {
                vpart += wss[WS_HV + c] / cnt;
                float denom = wss[WS_DENP + c] + wss[WS_CNT + c];
                dicep += 2.0f * wss[WS_INTER + c] / fmaxf(denom, 1e-7f);
            }
            regp += fast_sqrtf(wss[WS_MSUM + c] + 1e-12f);
        }
        float distp = 0.f;
        for (int idx = tid; idx < 39 * 39; idx += 256) {
            int i = idx / 39 + 1, j = idx % 39 + 1;
            if (i != j) {
                float dot = 0.f;
                for (int e = 0; e < EMB; ++e)
                    dot += wss[WS_MU + i * EMB + e] * wss[WS_MU + j * EMB + e];
                float md2 = wss[WS_MSUM + i] - 2.0f * dot + wss[WS_MSUM + j];
                float Md  = fast_sqrtf(fmaxf(md2, 0.0f) + 1e-12f);
                float h   = fmaxf(4.0f - Md, 0.0f);
                distp += h * h;
            }
        }
        atomicAdd(&red[0], vpart);
        atomicAdd(&red[2], distp);
        atomicAdd(&red[3], regp);
        atomicAdd(&red[4], dicep);
        __syncthreads();
        if (tid == 0) {
            float nbg   = fmaxf(wss[WS_CNT + 0], 1.0f);
            float var_t = red[0] / 39.0f;
            float push_t = wss[WS_PUSH] / nbg / 39.0f;
            float dist_t = red[2] / (39.0f * 38.0f);
            float reg_t  = red[3] / 40.0f;
            float inst_t = 1.0f - red[4] / 39.0f;
            total += var_t + dist_t + 0.001f * reg_t + inst_t + push_t;
        }
        __syncthreads();
    }
    if (tid == 0) out[0] = total / (float)BATCH;
}

extern "C" void kernel_launch(void* const* d_in, const int* in_sizes, int n_in,
                              void* d_out, int out_size, void* d_ws, size_t ws_size,
                              hipStream_t stream) {
    const float* in  = (const float*)d_in[0];
    const int*   tgt = (const int*)d_in[1];
    float* out = (float*)d_out;
    float* ws  = (float*)d_ws;

    const int nz = BATCH * WS_STRIDE;
    k_zero <<<(nz + 255) / 256, 256, 0, stream>>>(ws, nz);
    k_sums <<<BATCH * (PIX / PIXB), 256, 0, stream>>>(in, tgt, ws);
    k_means<<<BATCH, 256, 0, stream>>>(ws);
    k_main <<<BATCH * (PIX / BLK_PIX), 256, 0, stream>>>(in, tgt, ws);
    k_final<<<1, 256, 0, stream>>>(ws, out);
}